// BaseModel_42743514530330
// MI455X (gfx1250) — compile-verified
//
#include <hip/hip_runtime.h>
#include <stdint.h>

// Match the JAX reference's unfused mul/add sequence exactly: a 1-ulp IoU
// difference from FMA contraction could flip the 0.5 threshold or an argmax tie.
#pragma clang fp contract(off)

// Sizes from the reference: A=100000 pred boxes, B=2000 gt boxes, 64 images.
#define MAX_GT 2000

typedef uint32_t v4u __attribute__((ext_vector_type(4)));
typedef int      v8i __attribute__((ext_vector_type(8)));
typedef int      v4i __attribute__((ext_vector_type(4)));

struct GtSmem {
    float4 boxes[MAX_GT];   // 32 KB  (x1,y1,x2,y2 per gt box)
    int    img[MAX_GT];     //  8 KB  (sorted image index per gt box)
    int    lbl[MAX_GT];     //  8 KB  (class label per gt box)
};                          // 48 KB total per workgroup (<< 320 KB WGP LDS)

// Issue one TDM DMA: ndw contiguous DWORDs from global -> LDS.
// D# layout per CDNA5 ISA (group0 128b, group1 256b):
//   group0: [1:0]=count=1, [63:32]=lds_addr, [120:64]=global_addr, [127:126]=type=2
//   group1: [17:16]=data_size(2 => 4B), tensor_dim0=ndw, tensor_dim1=1,
//           tile_dim0=ndw (<=65535), tile_dim1=1, strides=ndw; no pad/iterate/mask.
__device__ __forceinline__ void tdm_load_dwords(uint32_t lds_off, uint64_t gaddr,
                                                uint32_t ndw) {
    v4u g0;
    g0.x = 1u;                                                   // count=1 (user)
    g0.y = lds_off;                                              // LDS byte address
    g0.z = (uint32_t)gaddr;                                      // global addr [31:0]
    g0.w = ((uint32_t)(gaddr >> 32) & 0x01FFFFFFu) | (2u << 30); // addr[56:32] | type=2
    v8i g1;
    g1[0] = (int)(2u << 16);                 // workgroup_mask=0, data_size=4B, no flags
    g1[1] = (int)((ndw & 0xFFFFu) << 16);    // abar_addr=0 | tensor_dim0[15:0]
    g1[2] = (int)((ndw >> 16) | (1u << 16)); // tensor_dim0[31:16] | tensor_dim1[15:0]=1
    g1[3] = (int)(ndw << 16);                // tensor_dim1[31:16]=0 | tile_dim0=ndw
    g1[4] = 1;                               // tile_dim1=1, tile_dim2=0(unused)
    g1[5] = (int)ndw;                        // tensor_dim0_stride[31:0]
    g1[6] = (int)((ndw & 0xFFFFu) << 16);    // stride0[47:32]=0 | stride1[15:0]
    g1[7] = (int)(ndw >> 16);                // stride1[47:16]
    v4i g2 = {0, 0, 0, 0};                   // 2D tensor: dims 2..4 unused
    v4i g3 = {0, 0, 0, 0};
    v8i g4 = {0, 0, 0, 0, 0, 0, 0, 0};       // extra group in 6-arg builtin: zeros
    __builtin_amdgcn_tensor_load_to_lds(g0, g1, g2, g3, g4, 0);
}

__global__ void __launch_bounds__(256)
assign_labels_kernel(const float* __restrict__ boxes_pred,
                     const float* __restrict__ boxes_gt,
                     const int*   __restrict__ gt_labels,
                     const int*   __restrict__ im_inds_pred,
                     const int*   __restrict__ im_inds_gt,
                     float*       __restrict__ out,
                     int A, int B) {
    __shared__ GtSmem s;

    // One wave per workgroup issues the three TDM descriptors (TDM ignores EXEC).
    if (threadIdx.x == 0) {
        uint32_t lds_boxes = (uint32_t)(size_t)(&s.boxes[0]);
        uint32_t lds_img   = (uint32_t)(size_t)(&s.img[0]);
        uint32_t lds_lbl   = (uint32_t)(size_t)(&s.lbl[0]);
        tdm_load_dwords(lds_boxes, (uint64_t)(size_t)boxes_gt,   (uint32_t)(4 * B));
        tdm_load_dwords(lds_img,   (uint64_t)(size_t)im_inds_gt, (uint32_t)B);
        tdm_load_dwords(lds_lbl,   (uint64_t)(size_t)gt_labels,  (uint32_t)B);
    }
    __builtin_amdgcn_s_wait_tensorcnt(0);  // issuing wave drains TENSORcnt; NOP for others
    __syncthreads();                       // publish LDS to all 8 waves

    int tid = blockIdx.x * blockDim.x + threadIdx.x;
    if (tid >= A) return;

    float4 p  = ((const float4*)boxes_pred)[tid];   // global_load_b128
    int   img = im_inds_pred[tid];
    float ap  = (p.z - p.x + 1.0f) * (p.w - p.y + 1.0f);

    // im_inds_gt is sorted: binary search the [start,end) range of this image.
    int lo = 0, hi = B;
    while (lo < hi) { int m = (lo + hi) >> 1; if (s.img[m] <  img) lo = m + 1; else hi = m; }
    int start = lo;
    hi = B;
    while (lo < hi) { int m = (lo + hi) >> 1; if (s.img[m] <= img) lo = m + 1; else hi = m; }
    int end = lo;

    // Masked row max/argmax; strict '>' keeps first-index-wins like jnp.argmax.
    float best = 0.0f;
    int   bestj = 0;
    for (int j = start; j < end; ++j) {
        float4 g  = s.boxes[j];                      // ds_load_b128
        float iw  = fminf(p.z, g.z) - fmaxf(p.x, g.x) + 1.0f;
        float ih  = fminf(p.w, g.w) - fmaxf(p.y, g.y) + 1.0f;
        iw = fmaxf(iw, 0.0f);
        ih = fmaxf(ih, 0.0f);
        float inter = iw * ih;
        float ag    = (g.z - g.x + 1.0f) * (g.w - g.y + 1.0f);
        float iou   = inter / (ap + ag - inter);
        if (iou > best) { best = iou; bestj = j; }
    }

    int label = (best >= 0.5f) ? s.lbl[bestj] : 0;   // empty range -> best=0 -> label 0
    out[tid] = (float)label;                         // labels 0..80 exact in fp32
}

extern "C" void kernel_launch(void* const* d_in, const int* in_sizes, int n_in,
                              void* d_out, int out_size, void* d_ws, size_t ws_size,
                              hipStream_t stream) {
    (void)n_in; (void)out_size; (void)d_ws; (void)ws_size;
    const float* boxes_pred   = (const float*)d_in[0];
    const float* boxes_gt     = (const float*)d_in[1];
    const int*   gt_labels    = (const int*)d_in[2];
    const int*   im_inds_pred = (const int*)d_in[3];
    const int*   im_inds_gt   = (const int*)d_in[4];

    int A = in_sizes[0] / 4;   // 100000
    int B = in_sizes[1] / 4;   // 2000 (must be <= MAX_GT)

    const int block = 256;     // 8 wave32 waves per workgroup
    const int grid  = (A + block - 1) / block;
    assign_labels_kernel<<<grid, block, 0, stream>>>(
        boxes_pred, boxes_gt, gt_labels, im_inds_pred, im_inds_gt,
        (float*)d_out, A, B);
}